// GranularComputing_39616778338801
// MI455X (gfx1250) — compile-verified
//
#include <hip/hip_runtime.h>
#include <hip/hip_bf16.h>

typedef float v2f __attribute__((ext_vector_type(2)));
typedef float v8f __attribute__((ext_vector_type(8)));

#define D64 64
#define CL 64
#define ROWS_PER_CHUNK 64
#define TPB 128
#define LSTR 68   // LDS row stride: 68*4B = 272B, 16B-aligned, bank stride 4 -> conflict-free

// ---------------------------------------------------------------------------
// fp32 WMMA: D(16x16,f32) += A(16x4,f32) x B(4x16,f32)
// emits v_wmma_f32_16x16x4_f32 (CDNA5 VOP3P)
// ---------------------------------------------------------------------------
__device__ __forceinline__ v8f wmma4(v2f a, v2f b, v8f c) {
  return __builtin_amdgcn_wmma_f32_16x16x4_f32(
      /*neg_a=*/false, a, /*neg_b=*/false, b,
      /*c_mod=*/(short)0, c, /*reuse_a=*/false, /*reuse_b=*/false);
}

// ===========================================================================
// Fused FCM iteration pass:
//   given C (64x64) and c2[j]=||C_j||^2, stream rows of X:
//     S = X @ C^T (WMMA), d = max(x2 + c2 - 2S, 0), w = 1/d,
//     U = w / rowsum(w), Um = U^2,
//     accumulate numPart += Um^T @ X (WMMA), denPart += colsum(Um)
// ===========================================================================
__global__ __launch_bounds__(TPB) void fcm_pass(
    const float* __restrict__ X, const float* __restrict__ x2g,
    const float* __restrict__ Ccur, const float* __restrict__ c2g,
    float* __restrict__ numPart, float* __restrict__ denPart,
    int N, int nChunks)
{
  __shared__ __align__(16) float C_lds[CL][LSTR];       // also reused as num staging
  __shared__ __align__(16) float X_lds[4][16][LSTR];
  __shared__ __align__(16) float Um_lds[4][16][LSTR];
  __shared__ float c2_lds[CL];
  __shared__ float den_lds[4][CL];

  const int tid   = threadIdx.x;
  const int wv    = tid >> 5;
  const int lane  = tid & 31;
  const int l16   = lane & 15;
  const int lhalf = lane >> 4;          // 0: lanes 0-15, 1: lanes 16-31

  for (int idx = tid; idx < CL * D64; idx += TPB)
    C_lds[idx >> 6][idx & 63] = Ccur[idx];
  if (tid < CL) c2_lds[tid] = c2g[tid];
  __syncthreads();

  v8f acc2[4][4];                       // 64x64 fp32 center-numerator accumulator
  #pragma unroll
  for (int a = 0; a < 4; ++a)
    #pragma unroll
    for (int b = 0; b < 4; ++b) acc2[a][b] = {};
  float denacc[4] = {0.f, 0.f, 0.f, 0.f};

  for (int chunk = blockIdx.x; chunk < nChunks; chunk += gridDim.x) {
    const int r0 = chunk * ROWS_PER_CHUNK + wv * 16;   // wave's first row

    { // prefetch next chunk of X into L2 (global_prefetch_b8)
      int nxt = chunk + gridDim.x;
      if (nxt < nChunks)
        __builtin_prefetch(X + (size_t)nxt * ROWS_PER_CHUNK * D64, 0, 1);
    }

    // ---- load this wave's 16x64 X tile into LDS (float4, coalesced) ----
    #pragma unroll
    for (int t = 0; t < 8; ++t) {
      int idx = lane + t * 32;
      int row = idx >> 4, c4 = (idx & 15) << 2;
      int g = r0 + row;
      float4 v = make_float4(0.f, 0.f, 0.f, 0.f);
      if (g < N) v = *(const float4*)(X + (size_t)g * D64 + c4);
      *(float4*)&X_lds[wv][row][c4] = v;
    }
    __syncthreads();

    // ---- S = X @ C^T : 4 cluster-tiles, K=64 -> 16 WMMAs each ----
    v8f sacc[4];
    #pragma unroll
    for (int jt = 0; jt < 4; ++jt) sacc[jt] = {};
    #pragma unroll
    for (int jt = 0; jt < 4; ++jt) {
      #pragma unroll
      for (int kk = 0; kk < D64; kk += 4) {
        v2f a, b;
        // A[i][k] = X[i][k]   (A layout: M=l16, K = kk + 2*lhalf + {0,1})
        a.x = X_lds[wv][l16][kk + 2 * lhalf];
        a.y = X_lds[wv][l16][kk + 2 * lhalf + 1];
        // B[k][j] = C[j][k]   (B layout: N=l16, K = kk + 2*lhalf + {0,1})
        b.x = C_lds[jt * 16 + l16][kk + 2 * lhalf];
        b.y = C_lds[jt * 16 + l16][kk + 2 * lhalf + 1];
        sacc[jt] = wmma4(a, b, sacc[jt]);
      }
    }

    // ---- membership update (m=2 => w = 1/d^2, Um = U^2) ----
    float x2r[8];
    bool  valid[8];
    #pragma unroll
    for (int r = 0; r < 8; ++r) {
      int g = r0 + r + lhalf * 8;       // D-layout: VGPR r -> row r (+8 for hi half)
      valid[r] = (g < N);
      x2r[r] = valid[r] ? x2g[g] : 0.f;
    }
    float w[4][8];
    #pragma unroll
    for (int jt = 0; jt < 4; ++jt) {
      float c2l = c2_lds[jt * 16 + l16];
      #pragma unroll
      for (int r = 0; r < 8; ++r) {
        float dsq = fmaxf(x2r[r] + c2l - 2.f * sacc[jt][r], 0.f);
        w[jt][r] = 1.f / fmaxf(dsq, 1e-20f);
      }
    }
    #pragma unroll
    for (int r = 0; r < 8; ++r) {
      float s = w[0][r] + w[1][r] + w[2][r] + w[3][r];
      #pragma unroll
      for (int m = 1; m < 16; m <<= 1) s += __shfl_xor(s, m, 32); // row sum over 64 j
      float inv = 1.f / s;
      #pragma unroll
      for (int jt = 0; jt < 4; ++jt) {
        float u  = w[jt][r] * inv;
        float um = u * u;
        w[jt][r] = valid[r] ? um : 0.f;
      }
    }

    // den[j] += sum_i Um[i][j]  (xor-16 folds the two row-halves together)
    #pragma unroll
    for (int jt = 0; jt < 4; ++jt) {
      float d = 0.f;
      #pragma unroll
      for (int r = 0; r < 8; ++r) d += w[jt][r];
      d += __shfl_xor(d, 16, 32);
      denacc[jt] += d;
    }

    // stage Um into LDS so it can be re-read in WMMA A-layout
    #pragma unroll
    for (int jt = 0; jt < 4; ++jt)
      #pragma unroll
      for (int r = 0; r < 8; ++r)
        Um_lds[wv][r + lhalf * 8][jt * 16 + l16] = w[jt][r];
    __syncthreads();

    // ---- num += Um^T @ X : 4x4 output tiles, K=16 -> 64 WMMAs ----
    #pragma unroll
    for (int jt = 0; jt < 4; ++jt)
      #pragma unroll
      for (int dt = 0; dt < 4; ++dt)
        #pragma unroll
        for (int kk = 0; kk < 16; kk += 4) {
          v2f a, b;
          // A[j][k] = Um[row k][cluster j]
          a.x = Um_lds[wv][kk + 2 * lhalf][jt * 16 + l16];
          a.y = Um_lds[wv][kk + 2 * lhalf + 1][jt * 16 + l16];
          // B[k][d] = X[row k][d]
          b.x = X_lds[wv][kk + 2 * lhalf][dt * 16 + l16];
          b.y = X_lds[wv][kk + 2 * lhalf + 1][dt * 16 + l16];
          acc2[jt][dt] = wmma4(a, b, acc2[jt][dt]);
        }
    __syncthreads();
  }

  // ---- block epilogue: deterministic partials to workspace ----
  if (lhalf == 0) {
    #pragma unroll
    for (int jt = 0; jt < 4; ++jt) den_lds[wv][jt * 16 + l16] = denacc[jt];
  }
  __syncthreads();
  if (tid < CL)
    denPart[(size_t)blockIdx.x * CL + tid] =
        den_lds[0][tid] + den_lds[1][tid] + den_lds[2][tid] + den_lds[3][tid];

  float* stage = &C_lds[0][0];          // 64*68 = 4352 floats >= 4*1024
  for (int jt = 0; jt < 4; ++jt) {
    __syncthreads();
    #pragma unroll
    for (int dt = 0; dt < 4; ++dt)
      #pragma unroll
      for (int r = 0; r < 8; ++r)
        stage[wv * 1024 + (r + lhalf * 8) * 64 + dt * 16 + l16] = acc2[jt][dt][r];
    __syncthreads();
    for (int idx = tid; idx < 1024; idx += TPB) {
      float s = stage[idx] + stage[1024 + idx] + stage[2048 + idx] + stage[3072 + idx];
      numPart[(size_t)blockIdx.x * 4096 + jt * 1024 + idx] = s;
    }
  }
}

// ===========================================================================
// Initial pass: U0 = rownorm(u0_raw); Um = U0^2; partials for C1 = center(U0);
// also precompute x2[i] = ||x_i||^2 (iteration-invariant).
// ===========================================================================
__global__ __launch_bounds__(TPB) void fcm_init(
    const float* __restrict__ X, const float* __restrict__ U0raw,
    float* __restrict__ x2g,
    float* __restrict__ numPart, float* __restrict__ denPart,
    int N, int nChunks)
{
  __shared__ __align__(16) float X_lds[4][16][LSTR];
  __shared__ __align__(16) float Um_lds[4][16][LSTR];
  __shared__ float stage[4][1024];
  __shared__ float inv_lds[4][16];
  __shared__ float den_lds[4][CL];

  const int tid   = threadIdx.x;
  const int wv    = tid >> 5;
  const int lane  = tid & 31;
  const int l16   = lane & 15;
  const int lhalf = lane >> 4;

  v8f acc2[4][4];
  #pragma unroll
  for (int a = 0; a < 4; ++a)
    #pragma unroll
    for (int b = 0; b < 4; ++b) acc2[a][b] = {};
  float denacc[4] = {0.f, 0.f, 0.f, 0.f};

  for (int chunk = blockIdx.x; chunk < nChunks; chunk += gridDim.x) {
    const int r0 = chunk * ROWS_PER_CHUNK + wv * 16;

    #pragma unroll
    for (int t = 0; t < 8; ++t) {
      int idx = lane + t * 32;
      int row = idx >> 4, c4 = (idx & 15) << 2;
      int g = r0 + row;
      float4 xv = make_float4(0.f, 0.f, 0.f, 0.f);
      float4 uv = make_float4(0.f, 0.f, 0.f, 0.f);
      if (g < N) {
        xv = *(const float4*)(X + (size_t)g * D64 + c4);
        uv = *(const float4*)(U0raw + (size_t)g * CL + c4);
      }
      *(float4*)&X_lds[wv][row][c4]  = xv;
      *(float4*)&Um_lds[wv][row][c4] = uv;   // raw u0 for now
    }
    __syncthreads();

    // row sums of u0 (for normalization) and of x^2 (for x2g)
    {
      float su = 0.f, sx = 0.f;
      for (int j = 0; j < 32; ++j) {
        float u = Um_lds[wv][l16][lhalf * 32 + j];
        float x = X_lds[wv][l16][lhalf * 32 + j];
        su += u;
        sx += x * x;
      }
      su += __shfl_xor(su, 16, 32);
      sx += __shfl_xor(sx, 16, 32);
      if (lhalf == 0) {
        inv_lds[wv][l16] = 1.f / su;
        int g = r0 + l16;
        if (g < N) x2g[g] = sx;
      }
    }
    __syncthreads();

    // Um = (u0 / rowsum)^2, zero for padding rows
    for (int idx = lane; idx < 1024; idx += 32) {
      int row = idx >> 6, j = idx & 63;
      float u  = Um_lds[wv][row][j] * inv_lds[wv][row];
      float um = u * u;
      if (r0 + row >= N) um = 0.f;
      Um_lds[wv][row][j] = um;
    }
    __syncthreads();

    // den[j] += colsum(Um)
    #pragma unroll
    for (int jt = 0; jt < 4; ++jt) {
      float d = 0.f;
      #pragma unroll
      for (int r = 0; r < 8; ++r) d += Um_lds[wv][r + lhalf * 8][jt * 16 + l16];
      d += __shfl_xor(d, 16, 32);
      denacc[jt] += d;
    }

    // num += Um^T @ X
    #pragma unroll
    for (int jt = 0; jt < 4; ++jt)
      #pragma unroll
      for (int dt = 0; dt < 4; ++dt)
        #pragma unroll
        for (int kk = 0; kk < 16; kk += 4) {
          v2f a, b;
          a.x = Um_lds[wv][kk + 2 * lhalf][jt * 16 + l16];
          a.y = Um_lds[wv][kk + 2 * lhalf + 1][jt * 16 + l16];
          b.x = X_lds[wv][kk + 2 * lhalf][dt * 16 + l16];
          b.y = X_lds[wv][kk + 2 * lhalf + 1][dt * 16 + l16];
          acc2[jt][dt] = wmma4(a, b, acc2[jt][dt]);
        }
    __syncthreads();
  }

  if (lhalf == 0) {
    #pragma unroll
    for (int jt = 0; jt < 4; ++jt) den_lds[wv][jt * 16 + l16] = denacc[jt];
  }
  __syncthreads();
  if (tid < CL)
    denPart[(size_t)blockIdx.x * CL + tid] =
        den_lds[0][tid] + den_lds[1][tid] + den_lds[2][tid] + den_lds[3][tid];

  for (int jt = 0; jt < 4; ++jt) {
    __syncthreads();
    #pragma unroll
    for (int dt = 0; dt < 4; ++dt)
      #pragma unroll
      for (int r = 0; r < 8; ++r)
        stage[wv][(r + lhalf * 8) * 64 + dt * 16 + l16] = acc2[jt][dt][r];
    __syncthreads();
    for (int idx = tid; idx < 1024; idx += TPB) {
      float s = stage[0][idx] + stage[1][idx] + stage[2][idx] + stage[3][idx];
      numPart[(size_t)blockIdx.x * 4096 + jt * 1024 + idx] = s;
    }
  }
}

// ===========================================================================
// Deterministic fixed-order reduction of block partials -> Cnext = num/den
// ===========================================================================
__global__ __launch_bounds__(256) void fcm_reduce(
    const float* __restrict__ numPart, const float* __restrict__ denPart,
    float* __restrict__ Cnext, int nblk)
{
  int idx = blockIdx.x * 256 + threadIdx.x;   // 0..4095 -> (j, d)
  int j = idx >> 6;
  float num = 0.f, den = 0.f;
  for (int b = 0; b < nblk; ++b) {
    num += numPart[(size_t)b * 4096 + idx];
    den += denPart[(size_t)b * 64 + j];
  }
  Cnext[idx] = num / fmaxf(den, 1e-30f);
}

// ===========================================================================
// Scan-step bookkeeping: diff = ||Cnext - Ccur||_F; freeze logic; c2 refresh.
//   if (!done_prev): V = Ccur; Ccur = Cnext; done |= (diff < EPS)
// ===========================================================================
__global__ __launch_bounds__(256) void fcm_update(
    const float* __restrict__ Cnext, float* __restrict__ Ccur,
    float* __restrict__ c2, float* __restrict__ V,
    float* __restrict__ flags, int first)
{
  __shared__ float red[256];
  const int tid = threadIdx.x;

  float local = 0.f;
  if (!first) {
    for (int idx = tid; idx < 4096; idx += 256) {
      float dlt = Cnext[idx] - Ccur[idx];
      local += dlt * dlt;
    }
  }
  red[tid] = local;
  __syncthreads();
  for (int s = 128; s > 0; s >>= 1) {
    if (tid < s) red[tid] += red[tid + s];
    __syncthreads();
  }
  const bool done_prev = (!first) && (flags[0] > 0.5f);
  const float diff = sqrtf(red[0]);
  const bool upd = first || !done_prev;
  __syncthreads();

  if (upd) {
    for (int idx = tid; idx < 4096; idx += 256) {
      float cn = Cnext[idx];
      if (!first) V[idx] = Ccur[idx];   // V gets the pre-update center
      Ccur[idx] = cn;
    }
  }
  __syncthreads();
  if (upd && tid < 64) {
    float s = 0.f;
    for (int d = 0; d < 64; ++d) {
      float v = Ccur[tid * 64 + d];
      s += v * v;
    }
    c2[tid] = s;
  }
  if (tid == 0) {
    if (first) flags[0] = 0.f;
    else if (!done_prev && diff < 1e-8f) flags[0] = 1.f;
  }
}

// ===========================================================================
extern "C" void kernel_launch(void* const* d_in, const int* in_sizes, int n_in,
                              void* d_out, int out_size, void* d_ws, size_t ws_size,
                              hipStream_t stream) {
  (void)n_in; (void)out_size;
  const float* X     = (const float*)d_in[0];
  const float* U0raw = (const float*)d_in[1];
  const int N = in_sizes[0] / D64;
  const int nChunks = (N + ROWS_PER_CHUNK - 1) / ROWS_PER_CHUNK;

  // workspace layout (floats)
  float* ws = (float*)d_ws;
  size_t off = 0;
  float* x2    = ws + off; off += (size_t)N;
  float* Ccur  = ws + off; off += 4096;
  float* c2    = ws + off; off += 64;
  float* Cnext = ws + off; off += 4096;
  float* flags = ws + off; off += 16;

  long long avail = (long long)(ws_size / 4) - (long long)off;
  int nblk = (int)(avail / (4096 + 64));
  if (nblk > 256) nblk = 256;
  if (nblk < 1)   nblk = 1;
  float* denPart = ws + off; off += (size_t)nblk * 64;
  float* numPart = ws + off;

  float* V = (float*)d_out;

  // C1 = center(U0), plus x2 precompute
  fcm_init<<<nblk, TPB, 0, stream>>>(X, U0raw, x2, numPart, denPart, N, nChunks);
  fcm_reduce<<<16, 256, 0, stream>>>(numPart, denPart, Cnext, nblk);
  fcm_update<<<1, 256, 0, stream>>>(Cnext, Ccur, c2, V, flags, 1);

  // MAX_ITER + 1 = 26 scan steps
  for (int it = 0; it < 26; ++it) {
    fcm_pass<<<nblk, TPB, 0, stream>>>(X, x2, Ccur, c2, numPart, denPart, N, nChunks);
    fcm_reduce<<<16, 256, 0, stream>>>(numPart, denPart, Cnext, nblk);
    fcm_update<<<1, 256, 0, stream>>>(Cnext, Ccur, c2, V, flags, 0);
  }
}